// VanillaRNN_84825604096388
// MI455X (gfx1250) — compile-verified
//
#include <hip/hip_runtime.h>
#include <cstdint>

// ---------------------------------------------------------------------------
// VanillaRNN (with the reference's rec = V[l] @ h[l, batch=l] broadcast bug)
// restructured for MI455X:
//   1) Y0small = X[b=0,1] @ W0^T            (small WMMA GEMM)
//   2) sequential 512-step recurrence on 3 H-vectors (grid-barrier kernel)
//   3) H0  = tanh(X  @ W0^T + Rec0[t] + b0) (big WMMA GEMM, fused epilogue)
//   4) Out = tanh(H0 @ W1^T + Rec1[t] + b1) (big WMMA GEMM, fused epilogue)
//   5) h_final = slices at t=S-1
// GEMMs: TDM tensor_load_to_lds (double buffered, s_wait_tensorcnt) feeding
// v_wmma_f32_16x16x4_f32. LDS tiles padded 1 dword / 16 (pitch 17) via the
// TDM pad fields -> bank-conflict-free fragment reads.
// ---------------------------------------------------------------------------

#define HID   1024
#define SEQL  512
#define NBAT  64

typedef float v2f  __attribute__((ext_vector_type(2)));
typedef float v8f  __attribute__((ext_vector_type(8)));
typedef unsigned int u32x4 __attribute__((ext_vector_type(4)));
typedef int   i32x4 __attribute__((ext_vector_type(4)));
typedef int   i32x8 __attribute__((ext_vector_type(8)));

// Issue one TDM 2D tile load: rows x 16 f32, row stride HID elems, into LDS
// at absolute byte address ldsByte, with 1-dword pad every 16 dwords.
__device__ __forceinline__ void tdm_load_tile16(const float* gsrc,
                                                unsigned ldsByte,
                                                unsigned rows) {
  unsigned long long ga = (unsigned long long)(uintptr_t)gsrc;
  u32x4 g0;
  g0[0] = 1u;                                   // count=1 (valid descriptor)
  g0[1] = ldsByte;                              // lds_addr
  g0[2] = (unsigned)ga;                         // global_addr[31:0]
  g0[3] = ((unsigned)(ga >> 32) & 0x01FFFFFFu)  // global_addr[56:32]
          | (2u << 30);                         // type = 2 (image/tensor)
  i32x8 g1;
  // data_size=2 (4B), pad_enable=1, pad_interval=3 (16 dwords), pad_amount=0 (1 dword)
  g1[0] = (int)((2u << 16) | (1u << 20) | (3u << 22));
  g1[1] = (int)(16u << 16);            // tensor_dim0 = 16 (tile always in-bounds)
  g1[2] = (int)(rows << 16);           // tensor_dim1 = rows
  g1[3] = (int)(16u << 16);            // tile_dim0 = 16
  g1[4] = (int)rows;                   // tile_dim1 = rows, tile_dim2 = 0
  g1[5] = (int)HID;                    // tensor_dim0_stride = 1024 elems
  g1[6] = 0;
  g1[7] = 0;
  i32x4 zz = {0, 0, 0, 0};
  i32x8 z8 = {0, 0, 0, 0, 0, 0, 0, 0};
  __builtin_amdgcn_tensor_load_to_lds(g0, g1, zz, zz, z8, 0);
}

// C[M,N] = A[M,K=1024] * Bw[N,K=1024]^T ; fuse: C = tanh(C + rec[m&511] + bias)
// Block tile 128M x 64N, 8 waves (4x2), each wave 32M x 32N (2x2 WMMA tiles).
__global__ __launch_bounds__(256)
void rnn_gemm_wmma(const float* __restrict__ A, const float* __restrict__ Bw,
                   float* __restrict__ C, const float* __restrict__ rec,
                   const float* __restrict__ bias, int fuse) {
  extern __shared__ float lds[];
  const unsigned ldsBase = __builtin_amdgcn_groupstaticsize();
  // dyn-LDS dword offsets: A buffers 128*17, B buffers 64*17, double buffered
  // A0=0  B0=2176  A1=3264  B1=5440   (total 6528 dwords = 26112 B)

  const int lane = threadIdx.x & 31;
  const int wid  = threadIdx.x >> 5;
  const int wm   = wid >> 1;            // 0..3 : M wave
  const int wn   = wid & 1;             // 0..1 : N wave
  const int r    = lane & 15;
  const int kh   = lane >> 4;           // 0/1 : which K/M half per ISA layout

  const int gm = blockIdx.x * 128;
  const int gn = blockIdx.y * 64;

  v8f acc00 = {0,0,0,0,0,0,0,0};
  v8f acc01 = {0,0,0,0,0,0,0,0};
  v8f acc10 = {0,0,0,0,0,0,0,0};
  v8f acc11 = {0,0,0,0,0,0,0,0};

  if (wid == 0) {                        // prologue: fill buffer 0
    tdm_load_tile16(A  + (size_t)gm * HID, ldsBase + 0u,     128u);
    tdm_load_tile16(Bw + (size_t)gn * HID, ldsBase + 8704u,  64u);
  }

  for (int ks = 0; ks < 64; ++ks) {      // K loop, 16 per step
    const int cur = ks & 1;
    if (wid == 0) {
      if (ks + 1 < 64) {
        const int nxt = cur ^ 1;
        tdm_load_tile16(A  + (size_t)gm * HID + (ks + 1) * 16,
                        ldsBase + (nxt ? 13056u : 0u), 128u);
        tdm_load_tile16(Bw + (size_t)gn * HID + (ks + 1) * 16,
                        ldsBase + (nxt ? 21760u : 8704u), 64u);
        __builtin_amdgcn_s_wait_tensorcnt((short)2);  // current buffer landed
      } else {
        __builtin_amdgcn_s_wait_tensorcnt((short)0);
      }
    }
    __syncthreads();

    const float* As = lds + (cur ? 3264 : 0);
    const float* Bs = lds + (cur ? 5440 : 2176);
    const int ra0 = (wm * 32 + r)      * 17;
    const int ra1 = (wm * 32 + 16 + r) * 17;
    const int rb0 = (wn * 32 + r)      * 17;
    const int rb1 = (wn * 32 + 16 + r) * 17;
#pragma unroll
    for (int kb = 0; kb < 16; kb += 4) {
      const int k0 = kb + 2 * kh;       // lanes 0-15: K=kb,kb+1; 16-31: kb+2,kb+3
      v2f a0, a1, b0, b1;
      a0.x = As[ra0 + k0]; a0.y = As[ra0 + k0 + 1];
      a1.x = As[ra1 + k0]; a1.y = As[ra1 + k0 + 1];
      b0.x = Bs[rb0 + k0]; b0.y = Bs[rb0 + k0 + 1];
      b1.x = Bs[rb1 + k0]; b1.y = Bs[rb1 + k0 + 1];
      acc00 = __builtin_amdgcn_wmma_f32_16x16x4_f32(false, a0, false, b0, (short)0, acc00, false, false);
      acc01 = __builtin_amdgcn_wmma_f32_16x16x4_f32(false, a0, false, b1, (short)0, acc01, false, false);
      acc10 = __builtin_amdgcn_wmma_f32_16x16x4_f32(false, a1, false, b0, (short)0, acc10, false, false);
      acc11 = __builtin_amdgcn_wmma_f32_16x16x4_f32(false, a1, false, b1, (short)0, acc11, false, false);
    }
    __syncthreads();                     // buffer reuse safety
  }

  // Epilogue. C/D layout: VGPR e -> rows e (lanes 0-15) / e+8 (lanes 16-31).
#pragma unroll
  for (int mi = 0; mi < 2; ++mi) {
#pragma unroll
    for (int ni = 0; ni < 2; ++ni) {
      v8f acc = (mi == 0) ? (ni == 0 ? acc00 : acc01)
                          : (ni == 0 ? acc10 : acc11);
      const int row0 = gm + wm * 32 + mi * 16 + kh * 8;
      const int col  = gn + wn * 32 + ni * 16 + r;
#pragma unroll
      for (int e = 0; e < 8; ++e) {
        const int row = row0 + e;
        float v = acc[e];
        if (fuse)
          v = tanhf(v + rec[(row & (SEQL - 1)) * HID + col] + bias[col]);
        C[(size_t)row * HID + col] = v;
      }
    }
  }
}

// ---- grid-wide barrier over d_ws counters (generation based, replay safe) --
__device__ __forceinline__ void grid_barrier(unsigned* cnt, unsigned* gen,
                                             unsigned nb) {
  __syncthreads();
  if (threadIdx.x == 0) {
    unsigned g = __hip_atomic_load(gen, __ATOMIC_RELAXED, __HIP_MEMORY_SCOPE_AGENT);
    __threadfence();
    unsigned arrived = atomicAdd(cnt, 1u);
    if (arrived == nb - 1) {
      __hip_atomic_store(cnt, 0u, __ATOMIC_RELAXED, __HIP_MEMORY_SCOPE_AGENT);
      __threadfence();
      atomicAdd(gen, 1u);
    } else {
      while (__hip_atomic_load(gen, __ATOMIC_RELAXED, __HIP_MEMORY_SCOPE_AGENT) == g)
        __builtin_amdgcn_s_sleep(2);
      __threadfence();
    }
  }
  __syncthreads();
}

__global__ void rnn_init(float* uvw, unsigned* cnt) {
  int i = blockIdx.x * blockDim.x + threadIdx.x;
  if (i < 6144) uvw[i] = 0.f;            // u/w/v double buffers
  if (i == 0) *cnt = 0u;                 // barrier arrival counter
}

// Sequential recurrence: u=h0[b0], w=h0[b1], v=h1[b1] (double buffered).
// Per step: Rec0[t]=V0*u, Rec1[t]=V1*v, then z=W1*w_t, v_t=tanh(z+Rec1+b1).
__global__ __launch_bounds__(256)
void rnn_seq(const float* __restrict__ W, const float* __restrict__ V,
             const float* __restrict__ b, const float* __restrict__ Y0s,
             float* __restrict__ Rec0, float* __restrict__ Rec1,
             float* __restrict__ uvw, unsigned* cnt, unsigned* gen) {
  const float* V0 = V;
  const float* V1 = V + 1048576;
  const float* W1 = W + 1048576;
  const float* b0 = b;
  const float* b1 = b + HID;
  float* ub = uvw;            // [2][1024]
  float* wb = uvw + 2048;
  float* vb = uvw + 4096;
  const unsigned NB = gridDim.x;
  const int tid = threadIdx.x, bid = blockIdx.x;

  for (int t = 0; t < SEQL; ++t) {
    const int curb = t & 1, prvb = curb ^ 1;
    // ---- phase 1: 2048 matvec rows (V0*u_{t-1}, V1*v_{t-1}) ----
    {
      const int d  = bid * 32 + (tid >> 3);
      const int l8 = tid & 7;
      if (d < 1024) {
        const float* Vr = V0 + (size_t)d * HID;
        const float* up = ub + prvb * HID;
        float acc = 0.f;
        for (int k = l8; k < HID; k += 8) acc += Vr[k] * up[k];
        acc += __shfl_xor(acc, 1); acc += __shfl_xor(acc, 2); acc += __shfl_xor(acc, 4);
        if (l8 == 0) {
          Rec0[t * HID + d] = acc;
          const float bb = b0[d];
          ub[curb * HID + d] = tanhf(Y0s[t * HID + d] + acc + bb);
          wb[curb * HID + d] = tanhf(Y0s[(SEQL + t) * HID + d] + acc + bb);
        }
      } else {
        const int row = d - 1024;
        const float* Vr = V1 + (size_t)row * HID;
        const float* vp = vb + prvb * HID;
        float acc = 0.f;
        for (int k = l8; k < HID; k += 8) acc += Vr[k] * vp[k];
        acc += __shfl_xor(acc, 1); acc += __shfl_xor(acc, 2); acc += __shfl_xor(acc, 4);
        if (l8 == 0) Rec1[t * HID + row] = acc;
      }
    }
    grid_barrier(cnt, gen, NB);
    // ---- phase 2: 1024 rows z = W1 * w_t ; v_t = tanh(z + Rec1 + b1) ----
    {
      const int d   = bid * 16 + (tid >> 4);
      const int l16 = tid & 15;
      const float* Wr = W1 + (size_t)d * HID;
      const float* wp = wb + curb * HID;
      float acc = 0.f;
      for (int k = l16; k < HID; k += 16) acc += Wr[k] * wp[k];
      acc += __shfl_xor(acc, 1); acc += __shfl_xor(acc, 2);
      acc += __shfl_xor(acc, 4); acc += __shfl_xor(acc, 8);
      if (l16 == 0)
        vb[curb * HID + d] = tanhf(acc + Rec1[t * HID + d] + b1[d]);
    }
    grid_barrier(cnt, gen, NB);
  }
}

__global__ void rnn_finals(const float* __restrict__ H0,
                           const float* __restrict__ Out,
                           float* __restrict__ hf) {
  int i = blockIdx.x * blockDim.x + threadIdx.x;   // 0..131071  [L,B,H]
  int layer = i >> 16;
  int bh    = i & 65535;
  int bb    = bh >> 10;
  int n     = bh & 1023;
  const float* src = layer ? Out : H0;
  hf[i] = src[((size_t)bb * SEQL + (SEQL - 1)) * HID + n];
}

extern "C" void kernel_launch(void* const* d_in, const int* in_sizes, int n_in,
                              void* d_out, int out_size, void* d_ws, size_t ws_size,
                              hipStream_t stream) {
  (void)in_sizes; (void)n_in; (void)out_size; (void)ws_size;
  const float* x = (const float*)d_in[0];   // [64,512,1024]
  const float* W = (const float*)d_in[1];   // [2,1024,1024]
  const float* V = (const float*)d_in[2];   // [2,1024,1024]
  const float* b = (const float*)d_in[3];   // [2,1024]
  float* out = (float*)d_out;                        // [64,512,1024]
  float* hf  = out + (size_t)NBAT * SEQL * HID;      // [2,64,1024]

  float* ws   = (float*)d_ws;               // needs ~142.6 MB
  float* H0   = ws;                         // 33,554,432 f
  float* Y0s  = ws + 33554432ull;           //  1,048,576 f
  float* Rec0 = ws + 34603008ull;           //    524,288 f
  float* Rec1 = ws + 35127296ull;           //    524,288 f
  float* uvw  = ws + 35651584ull;           //      6,144 f
  unsigned* cnt = (unsigned*)(ws + 35657728ull);
  unsigned* gen = cnt + 1;

  const size_t ldsBytes = 26112;            // double-buffered A(128x17)+B(64x17)

  // 1) Y0small = X[batch 0,1] @ W0^T   (rows 0..1023 of X, contiguous)
  rnn_gemm_wmma<<<dim3(8, 16), 256, ldsBytes, stream>>>(x, W, Y0s, Rec0, b, 0);
  // 2) zero u/w/v state + barrier counter
  rnn_init<<<24, 256, 0, stream>>>(uvw, cnt);
  // 3) sequential recurrence -> Rec0[t], Rec1[t]
  rnn_seq<<<64, 256, 0, stream>>>(W, V, b, Y0s, Rec0, Rec1, uvw, cnt, gen);
  // 4) H0 = tanh(X @ W0^T + Rec0 + b0)
  rnn_gemm_wmma<<<dim3(256, 16), 256, ldsBytes, stream>>>(x, W, H0, Rec0, b, 1);
  // 5) Out = tanh(H0 @ W1^T + Rec1 + b1)
  rnn_gemm_wmma<<<dim3(256, 16), 256, ldsBytes, stream>>>(H0, W + 1048576, out,
                                                          Rec1, b + HID, 1);
  // 6) h_final slices
  rnn_finals<<<512, 256, 0, stream>>>(H0, out, hf);
}